// GCN_84799834293080
// MI455X (gfx1250) — compile-verified
//
#include <hip/hip_runtime.h>

// GCN 2-layer forward for MI455X (gfx1250, wave32).
// GEMM1 (100000x128 @ 128x16) uses native fp32 WMMA v_wmma_f32_16x16x4_f32.
// Scatter passes are the memory-bound core: float L2 atomics + float4 gathers.
// Self-loop edges are folded algebraically (agg += dis[n]^2 * h[n]) so we never
// materialize the E+N concatenated edge list.

#define FIN  128
#define HID  16
#define NCLS 8

typedef float v2f __attribute__((ext_vector_type(2)));
typedef float v8f __attribute__((ext_vector_type(8)));

static __host__ int cdiv(long a, int b) { return (int)((a + b - 1) / b); }

// ---------------- zero workspace (deg/agg1/agg2) ----------------
__global__ void k_zero(float* __restrict__ p, long n) {
    long i = (long)blockIdx.x * blockDim.x + threadIdx.x;
    if (i < n) p[i] = 0.0f;
}

// ---------------- degree from dst counts ----------------
__global__ void k_deg(const int* __restrict__ dst, int E, float* __restrict__ deg) {
    int e = blockIdx.x * blockDim.x + threadIdx.x;
    if (e < E) atomicAdd(&deg[dst[e]], 1.0f);
}

// deg -> dis = rsqrt(deg + 1)   (+1 = self-loop; always > 0)
__global__ void k_dis(float* __restrict__ degdis, int N) {
    int n = blockIdx.x * blockDim.x + threadIdx.x;
    if (n < N) degdis[n] = rsqrtf(degdis[n] + 1.0f);
}

// ---------------- GEMM1: h1[N,16] = x[N,128] @ W1[128,16]  (fp32 WMMA) -------
// One wave per 16-row tile. 32 x v_wmma_f32_16x16x4_f32 accumulate K=128.
// A layout (ISA 7.12.2, 32-bit A 16x4): lane L -> row L&15, K = (L>>4)*2 + {0,1}
// B layout (4x16): lane L -> col L&15, same K pair.
// D layout (16x16 f32): VGPR i -> row (L>>4)*8 + i, col L&15.
__global__ void k_gemm1(const float* __restrict__ x, const float* __restrict__ W1,
                        float* __restrict__ h1, int N, int ntiles) {
    int wave = threadIdx.x >> 5;
    int lane = threadIdx.x & 31;
    int tile = blockIdx.x * (blockDim.x >> 5) + wave;
    if (tile >= ntiles) return;            // wave-uniform: EXEC all-ones below

    int m0   = tile * 16;
    int half = lane >> 4;
    int col  = lane & 15;
    int row  = m0 + (lane & 15);
    int lrow = row < N ? row : N - 1;      // clamp loads for a partial tail tile
    int kb   = half * 2;

    const float* xr = x + (size_t)lrow * FIN;
    v8f acc = {};
#pragma unroll
    for (int kk = 0; kk < FIN / 4; ++kk) {
        int k = kk * 4 + kb;
        v2f a = *(const v2f*)(xr + k);                 // x[row][k..k+1], 8B load
        v2f b;
        b[0] = W1[(size_t)k * HID + col];              // W1[k][col]
        b[1] = W1[(size_t)(k + 1) * HID + col];        // W1[k+1][col]
        acc = __builtin_amdgcn_wmma_f32_16x16x4_f32(
            /*neg_a=*/false, a, /*neg_b=*/false, b,
            /*c_mod=*/(short)0, acc, /*reuse_a=*/false, /*reuse_b=*/false);
    }
#pragma unroll
    for (int i = 0; i < 8; ++i) {
        int r = m0 + half * 8 + i;
        if (r < N) h1[(size_t)r * HID + col] = acc[i];
    }
}

// ---------------- normalized scatter-add over real edges --------------------
// FSHIFT: log2(# float4 groups per edge). Layer1: F=16 -> 2, Layer2: F=8 -> 1.
template <int FSHIFT>
__global__ void k_scatter(const int* __restrict__ src, const int* __restrict__ dst,
                          const float* __restrict__ dis, const float* __restrict__ hin,
                          float* __restrict__ agg, int E) {
    const int F4 = 1 << FSHIFT;            // float4 groups per edge
    const int F  = F4 * 4;                 // features per edge
    long t = (long)blockIdx.x * blockDim.x + threadIdx.x;
    if (t >= (long)E << FSHIFT) return;
    int e = (int)(t >> FSHIFT);
    int q = (int)(t & (F4 - 1));
    int s = src[e];
    int d = dst[e];
    float w = dis[s] * dis[d];
    float4 v = *(const float4*)(hin + (size_t)s * F + q * 4);
    float* o = agg + (size_t)d * F + q * 4;
    atomicAdd(o + 0, w * v.x);
    atomicAdd(o + 1, w * v.y);
    atomicAdd(o + 2, w * v.z);
    atomicAdd(o + 3, w * v.w);
}

// ---------------- self-loop + bias + ReLU + GEMM2 fused per node ------------
__global__ void k_layer2(const float* __restrict__ agg1, const float* __restrict__ h1,
                         const float* __restrict__ dis, const float* __restrict__ b1,
                         const float* __restrict__ W2, float* __restrict__ h2, int N) {
    int n = blockIdx.x * blockDim.x + threadIdx.x;
    if (n >= N) return;
    float sw = dis[n] * dis[n];            // self-loop weight dis[n]^2
    float t[HID];
    const float4* a4 = (const float4*)(agg1 + (size_t)n * HID);
    const float4* h4 = (const float4*)(h1 + (size_t)n * HID);
#pragma unroll
    for (int i = 0; i < 4; ++i) {
        float4 av = a4[i], hv = h4[i];
        t[4 * i + 0] = fmaxf(av.x + sw * hv.x + b1[4 * i + 0], 0.0f);
        t[4 * i + 1] = fmaxf(av.y + sw * hv.y + b1[4 * i + 1], 0.0f);
        t[4 * i + 2] = fmaxf(av.z + sw * hv.z + b1[4 * i + 2], 0.0f);
        t[4 * i + 3] = fmaxf(av.w + sw * hv.w + b1[4 * i + 3], 0.0f);
    }
    float* o = h2 + (size_t)n * NCLS;
#pragma unroll
    for (int j = 0; j < NCLS; ++j) {
        float s = 0.0f;
#pragma unroll
        for (int k = 0; k < HID; ++k) s += t[k] * W2[k * NCLS + j];   // uniform: s_loads
        o[j] = s;
    }
}

// ---------------- self-loop + bias + log_softmax ----------------------------
__global__ void k_out(const float* __restrict__ agg2, const float* __restrict__ h2,
                      const float* __restrict__ dis, const float* __restrict__ b2,
                      float* __restrict__ out, int N) {
    int n = blockIdx.x * blockDim.x + threadIdx.x;
    if (n >= N) return;
    float sw = dis[n] * dis[n];
    float z[NCLS];
    float m = -3.402823466e+38f;
#pragma unroll
    for (int j = 0; j < NCLS; ++j) {
        z[j] = agg2[(size_t)n * NCLS + j] + sw * h2[(size_t)n * NCLS + j] + b2[j];
        m = fmaxf(m, z[j]);
    }
    float s = 0.0f;
#pragma unroll
    for (int j = 0; j < NCLS; ++j) s += __expf(z[j] - m);
    float l = m + __logf(s);
#pragma unroll
    for (int j = 0; j < NCLS; ++j) out[(size_t)n * NCLS + j] = z[j] - l;
}

extern "C" void kernel_launch(void* const* d_in, const int* in_sizes, int n_in,
                              void* d_out, int out_size, void* d_ws, size_t ws_size,
                              hipStream_t stream) {
    const float* x  = (const float*)d_in[0];
    const int*   ei = (const int*)d_in[1];     // per harness: integer -> const int*
    const float* W1 = (const float*)d_in[2];
    const float* b1 = (const float*)d_in[3];
    const float* W2 = (const float*)d_in[4];
    const float* b2 = (const float*)d_in[5];
    float* out = (float*)d_out;

    const int N = in_sizes[0] / FIN;           // 100000
    const int E = in_sizes[1] / 2;             // 3200000
    const int* src = ei;
    const int* dst = ei + E;

    // workspace layout (floats): dis | agg1 | agg2 | h1 | h2
    float* ws   = (float*)d_ws;
    float* dis  = ws;                                  // N    (deg, then rsqrt)
    float* agg1 = dis + N;                             // 16N
    float* agg2 = agg1 + (size_t)N * HID;              // 8N
    float* h1   = agg2 + (size_t)N * NCLS;             // 16N
    float* h2   = h1 + (size_t)N * HID;                // 8N

    const int B = 256;
    const long nz = (long)N * (1 + HID + NCLS);        // dis+agg1+agg2 region

    k_zero<<<cdiv(nz, B), B, 0, stream>>>(ws, nz);
    k_deg<<<cdiv(E, B), B, 0, stream>>>(dst, E, dis);
    k_dis<<<cdiv(N, B), B, 0, stream>>>(dis, N);

    const int ntiles = (N + 15) / 16;                  // 6250
    k_gemm1<<<cdiv(ntiles, B / 32), B, 0, stream>>>(x, W1, h1, N, ntiles);

    k_scatter<2><<<cdiv((long)E * 4, B), B, 0, stream>>>(src, dst, dis, h1, agg1, E);
    k_layer2<<<cdiv(N, B), B, 0, stream>>>(agg1, h1, dis, b1, W2, h2, N);
    k_scatter<1><<<cdiv((long)E * 2, B), B, 0, stream>>>(src, dst, dis, h2, agg2, E);
    k_out<<<cdiv(N, B), B, 0, stream>>>(agg2, h2, dis, b2, out, N);
}